// BMeshRasterizer_10977936408785
// MI455X (gfx1250) — compile-verified
//
#include <hip/hip_runtime.h>

// Soft rasterizer for MI455X (gfx1250, wave32).
// Per-face linear-form coefficients + bboxes are precomputed into LDS once per
// 256-face chunk. Each wave (16-pixel tile) culls face-quads against its tile
// bbox via v_cmp + ballot (wave-uniform survivor masks, SALU bit-loop), then
// V_WMMA_F32_16X16X4_F32 evaluates {t0,t1,t2,pz_num} for 16 pixels x 4
// surviving faces per instruction. Each lane holds all four forms for 2 faces
// of its own pixel, so the perspective divide / inside test / top-8 insertion
// stay per-lane behind an execz-skippable branch.

typedef __attribute__((ext_vector_type(2))) float v2f;
typedef __attribute__((ext_vector_type(8))) float v8f;

#define IMG     256
#define NFACE   4096
#define KF      8
#define CHUNK   256
#define NCHUNK  (NFACE / CHUNK)
#define FS      28              // floats per precomputed face record
#define BBOFF   (CHUNK * FS)    // bbox region: [7168, 8192) floats
#define EPSR    1e-8f

__device__ __forceinline__ float seg_d2(float px, float py, float ax, float ay,
                                        float abx, float aby, float invL2) {
    float apx = px - ax, apy = py - ay;
    float t = __saturatef((apx * abx + apy * aby) * invL2);
    float dx = apx - t * abx, dy = apy - t * aby;
    return dx * dx + dy * dy;
}

__global__ __launch_bounds__(256)
void BMeshRasterizer_gfx1250_kernel(const float* __restrict__ fv,
                                    float* __restrict__ outF) {
    // Overlay: staging uses [0, 8192) (records + bboxes); merge uses [0, 12288).
    __shared__ float lds[12288];

    const int tid    = threadIdx.x;
    const int lane   = tid & 31;
    const int wave   = tid >> 5;
    const int laneLo = lane & 15;
    const int hiHalf = lane >> 4;        // 0: faces {0,1} mod 4 ; 1: faces {2,3}

    // Each block = 8 waves x 16 pixels = 128 contiguous pixels of one row.
    const int blk    = blockIdx.x;
    const int y      = blk >> 1;
    const int xbase0 = (blk & 1) * 128 + wave * 16;
    const int x      = xbase0 + laneLo;

    const float px = (255.0f - 2.0f * (float)x) * (1.0f / 256.0f);
    const float py = (255.0f - 2.0f * (float)y) * (1.0f / 256.0f);
    // Wave-uniform tile x-interval (px decreases with x).
    const float txhi = (255.0f - 2.0f * (float)xbase0) * (1.0f / 256.0f);
    const float txlo = (255.0f - 2.0f * (float)(xbase0 + 15)) * (1.0f / 256.0f);

    // B matrix (4x16, K=4): column n = pixel n of this wave's tile.
    // Lanes 0-15 hold K=0 (px), K=1 (py); lanes 16-31 hold K=2 (1), K=3 (0).
    v2f Bm;
    Bm.x = hiHalf ? 1.0f : px;
    Bm.y = hiHalf ? 0.0f : py;

    // Per-lane sorted top-8 (ascending z) over this half's face subset.
    float zb[KF], w0b[KF], w1b[KF], w2b[KF], sdb[KF];
    int   ib[KF];
#pragma unroll
    for (int k = 0; k < KF; ++k) {
        zb[k] = __builtin_inff();
        ib[k] = -1;
        w0b[k] = w1b[k] = w2b[k] = sdb[k] = -1.0f;
    }

    for (int c = 0; c < NCHUNK; ++c) {
        __syncthreads();
        { // ---- stage + precompute one face per thread into LDS ----
            const int f = c * CHUNK + tid;
            const float* p = fv + f * 9;
            if (c + 1 < NCHUNK)
                __builtin_prefetch(fv + (f + CHUNK) * 9, 0, 1);  // global_prefetch_b8

            float v0x = p[0], v0y = p[1], z0 = p[2];
            float v1x = p[3], v1y = p[4], z1 = p[5];
            float v2x = p[6], v2y = p[7], z2 = p[8];

            float area  = (v2x - v0x) * (v1y - v0y) - (v2y - v0y) * (v1x - v0x);
            float aab   = fabsf(area);
            float flag  = (aab > EPSR) ? 1.0f : 0.0f;
            float asafe = (aab < EPSR) ? EPSR : area;
            float inv   = flag / asafe;   // degenerate face -> all coeffs 0
            float s0 = z1 * z2 * inv, s1 = z0 * z2 * inv, s2 = z0 * z1 * inv;

            // ti(p) = Ai*px + Bi*py + Ci  (edge fn scaled by zj*zk/area)
            float A0 = (v2y - v1y) * s0, B0 = (v1x - v2x) * s0;
            float C0 = (v1y * (v2x - v1x) - v1x * (v2y - v1y)) * s0;
            float A1 = (v0y - v2y) * s1, B1 = (v2x - v0x) * s1;
            float C1 = (v2y * (v0x - v2x) - v2x * (v0y - v2y)) * s1;
            float A2 = (v1y - v0y) * s2, B2 = (v0x - v1x) * s2;
            float C2 = (v0y * (v1x - v0x) - v0x * (v1y - v0y)) * s2;
            // pz numerator form: t0*z0 + t1*z1 + t2*z2
            float PA = A0 * z0 + A1 * z1 + A2 * z2;
            float PB = B0 * z0 + B1 * z1 + B2 * z2;
            float PC = C0 * z0 + C1 * z1 + C2 * z2;

            float* q = &lds[tid * FS];
            q[0] = A0;  q[1] = B0;  q[2]  = C0;
            q[3] = A1;  q[4] = B1;  q[5]  = C1;
            q[6] = A2;  q[7] = B2;  q[8]  = C2;
            q[9] = PA;  q[10] = PB; q[11] = PC;
            // segment constants, packed for aligned float4 broadcast reads
            float e01x = v1x - v0x, e01y = v1y - v0y;
            float e12x = v2x - v1x, e12y = v2y - v1y;
            float e20x = v0x - v2x, e20y = v0y - v2y;
            q[12] = v0x; q[13] = v0y; q[14] = e01x; q[15] = e01y;
            q[16] = 1.0f / fmaxf(e01x * e01x + e01y * e01y, EPSR);
            q[17] = v1x; q[18] = v1y; q[19] = e12x;
            q[20] = e12y;
            q[21] = 1.0f / fmaxf(e12x * e12x + e12y * e12y, EPSR);
            q[22] = v2x; q[23] = v2y;
            q[24] = e20x; q[25] = e20y;
            q[26] = 1.0f / fmaxf(e20x * e20x + e20y * e20y, EPSR);
            q[27] = 0.0f;

            // face bbox for tile culling (aligned float4)
            float4 bb;
            bb.x = fminf(fminf(v0x, v1x), v2x);   // xlo
            bb.y = fmaxf(fmaxf(v0x, v1x), v2x);   // xhi
            bb.z = fminf(fminf(v0y, v1y), v2y);   // ylo
            bb.w = fmaxf(fmaxf(v0y, v1y), v2y);   // yhi
            *(float4*)&lds[BBOFF + tid * 4] = bb;
        }
        __syncthreads();

        // ---- per-wave quad cull: 64 quads -> two wave-uniform bitmasks ----
        uint32_t qmask[2];
#pragma unroll
        for (int h = 0; h < 2; ++h) {
            const int q = lane + h * 32;          // quad index 0..63
            bool ov = false;
#pragma unroll
            for (int u = 0; u < 4; ++u) {
                const float4 bb = *(const float4*)&lds[BBOFF + (q * 4 + u) * 4];
                ov |= (bb.y >= txlo) && (bb.x <= txhi) &&
                      (bb.w >= py)   && (bb.z <= py);
            }
            qmask[h] = __builtin_amdgcn_ballot_w32(ov);
        }

        // ---- surviving quads only: 4 faces per WMMA ----
#pragma unroll
        for (int h = 0; h < 2; ++h) {
            uint32_t m = qmask[h];
            while (m) {               // wave-uniform SALU bit loop, ascending
                const int fq = (h << 5) + __builtin_ctz(m);
                m &= m - 1;

                // A matrix (16x4): row r = face fq*4 + r/4, form r%4 -> [A,B,C,0]
                const int abase = (fq * 4 + (laneLo >> 2)) * FS
                                + (laneLo & 3) * 3 + hiHalf * 2;
                float a0 = lds[abase];
                float a1 = lds[abase + 1];      // hi half over-read (in-bounds)
                v2f Am;
                Am.x = a0;                      // lo: A ; hi: C
                Am.y = hiHalf ? 0.0f : a1;      // lo: B ; hi: 0

                v8f acc = {};
                acc = __builtin_amdgcn_wmma_f32_16x16x4_f32(
                    false, Am, false, Bm, (short)0, acc, false, false);

                // Lane now holds t0,t1,t2,pz_num for 2 faces of its pixel.
#pragma unroll
                for (int j = 0; j < 2; ++j) {
                    float t0 = acc[j * 4 + 0], t1 = acc[j * 4 + 1];
                    float t2 = acc[j * 4 + 2], pzn = acc[j * 4 + 3];
                    float den = t0 + t1 + t2;
                    den = (fabsf(den) < EPSR) ? EPSR : den;
                    float rden = __builtin_amdgcn_rcpf(den);   // v_rcp_f32
                    float w0 = t0 * rden, w1 = t1 * rden, w2 = t2 * rden;
                    float pz = pzn * rden;

                    bool inside = (w0 > 0.0f) && (w1 > 0.0f) && (w2 > 0.0f);
                    if (inside && (pz >= 0.0f) && (pz < zb[KF - 1])) {
                        const int fl = fq * 4 + hiHalf * 2 + j;
                        const float4 s0 = *(const float4*)&lds[fl * FS + 12];
                        const float4 s1 = *(const float4*)&lds[fl * FS + 16];
                        const float4 s2 = *(const float4*)&lds[fl * FS + 20];
                        const float4 s3 = *(const float4*)&lds[fl * FS + 24];
                        float da = seg_d2(px, py, s0.x, s0.y, s0.z, s0.w, s1.x);
                        float db = seg_d2(px, py, s1.y, s1.z, s1.w, s2.x, s2.y);
                        float dc = seg_d2(px, py, s2.z, s2.w, s3.x, s3.y, s3.z);
                        float sd = -fminf(da, fminf(db, dc));  // inside => -d2
                        const int fgi = c * CHUNK + fl;
#pragma unroll
                        for (int k = KF - 1; k >= 0; --k) {
                            bool ltp = (k > 0) ? (pz < zb[k - 1]) : false;
                            if (pz < zb[k]) {
                                zb[k]  = ltp ? zb[k - 1]  : pz;
                                ib[k]  = ltp ? ib[k - 1]  : fgi;
                                w0b[k] = ltp ? w0b[k - 1] : w0;
                                w1b[k] = ltp ? w1b[k - 1] : w1;
                                w2b[k] = ltp ? w2b[k - 1] : w2;
                                sdb[k] = ltp ? sdb[k - 1] : sd;
                            }
                        }
                    }
                }
            }
        }
    }

    // ---- merge the two half-lists per pixel via LDS, then write outputs ----
    __syncthreads();
    {
        float* m = &lds[(wave * 32 + lane) * (KF * 6)];
#pragma unroll
        for (int k = 0; k < KF; ++k) {
            m[k * 6 + 0] = zb[k];
            m[k * 6 + 1] = __int_as_float(ib[k]);
            m[k * 6 + 2] = w0b[k];
            m[k * 6 + 3] = w1b[k];
            m[k * 6 + 4] = w2b[k];
            m[k * 6 + 5] = sdb[k];
        }
    }
    __syncthreads();

    if (!hiHalf) {
        const float* la = &lds[(wave * 32 + laneLo) * (KF * 6)];
        const float* lb = &lds[(wave * 32 + 16 + laneLo) * (KF * 6)];
        int*   outI = (int*)outF;                              // p2f (int32)
        float* outZ = outF + IMG * IMG * KF;                   // zbuf
        float* outB = outF + 2 * IMG * IMG * KF;               // bary
        float* outD = outF + 2 * IMG * IMG * KF + IMG * IMG * KF * 3; // dists
        const int pbase = (y * IMG + x) * KF;

        int i = 0, j = 0;
#pragma unroll
        for (int k = 0; k < KF; ++k) {
            float za = la[i * 6], zB = lb[j * 6];
            bool ta = (za <= zB);
            const float* e = ta ? (la + i * 6) : (lb + j * 6);
            float z   = e[0];
            int   fid = __float_as_int(e[1]);
            float a0  = e[2], a1 = e[3], a2 = e[4], sd = e[5];
            bool hit = (z < 3.0e38f);
            const int o = pbase + k;
            outI[o]         = hit ? fid : -1;
            outZ[o]         = hit ? z   : -1.0f;
            outB[o * 3 + 0] = hit ? a0  : -1.0f;
            outB[o * 3 + 1] = hit ? a1  : -1.0f;
            outB[o * 3 + 2] = hit ? a2  : -1.0f;
            outD[o]         = hit ? sd  : -1.0f;
            i += ta ? 1 : 0;
            j += ta ? 0 : 1;
        }
    }
}

extern "C" void kernel_launch(void* const* d_in, const int* in_sizes, int n_in,
                              void* d_out, int out_size, void* d_ws, size_t ws_size,
                              hipStream_t stream) {
    (void)in_sizes; (void)n_in; (void)out_size; (void)d_ws; (void)ws_size;
    const float* fv = (const float*)d_in[0];   // face_verts: 4096 x 3 x 3 f32
    float* out = (float*)d_out;
    dim3 grid(IMG * IMG / 128);   // 512 blocks x 128 pixels
    dim3 block(256);              // 8 wave32 x 16-pixel WMMA tiles
    hipLaunchKernelGGL(BMeshRasterizer_gfx1250_kernel, grid, block, 0, stream,
                       fv, out);
}